// HDEmbedder_38809324486986
// MI455X (gfx1250) — compile-verified
//
#include <hip/hip_runtime.h>
#include <stdint.h>

// Problem constants (from reference)
#define NSAMP 4096
#define NCH   32
#define DDIM  2048
#define NLEV  100
#define DWPR  64                       // packed dwords per row: 2048 bits / 32
#define VAL_DW (NLEV * DWPR)           // 6400 dwords (25.6 KB)
#define POS_DW (NCH * DWPR)            // 2048 dwords (8 KB)
#define TOT_DW (VAL_DW + POS_DW)       // 8448 dwords (33.8 KB) -> fits LDS easily

typedef unsigned int v4u __attribute__((ext_vector_type(4)));
typedef int          v4i __attribute__((ext_vector_type(4)));
typedef int          v8i __attribute__((ext_vector_type(8)));

// ---------------------------------------------------------------------------
// Kernel 1: per-column min/max over the batch (coalesced: wave = 32 channels)
// ws[0..31] = min, ws[32..63] = max
// ---------------------------------------------------------------------------
__global__ __launch_bounds__(1024) void hd_minmax_kernel(
    const float* __restrict__ feat, float* __restrict__ ws) {
  __shared__ float smn[1024];
  __shared__ float smx[1024];
  const int tid = threadIdx.x;
  const int c = tid & 31;   // channel
  const int g = tid >> 5;   // row group 0..31
  float mn =  3.402823466e38f;
  float mx = -3.402823466e38f;
  for (int n = g; n < NSAMP; n += 32) {
    float v = feat[n * NCH + c];        // wave reads 32 consecutive floats
    mn = fminf(mn, v);
    mx = fmaxf(mx, v);
  }
  smn[tid] = mn;
  smx[tid] = mx;
  __syncthreads();
  if (g == 0) {
    for (int k = 1; k < 32; ++k) {
      mn = fminf(mn, smn[k * 32 + c]);
      mx = fmaxf(mx, smx[k * 32 + c]);
    }
    ws[c] = mn;
    ws[32 + c] = mx;
  }
}

// ---------------------------------------------------------------------------
// Kernel 2: pack sign bits of val_w (rows 0..99) and pos_w (rows 100..131)
// into bit-planes. For row r, packed dword index w64 = h*32 + w holds bits
// b=0..31 where bit b corresponds to d = h*1024 + b*32 + w.  (bit=1 <=> -1.0)
// This transposed packing makes the final output stores fully coalesced.
// ---------------------------------------------------------------------------
__global__ __launch_bounds__(256) void hd_pack_kernel(
    const float* __restrict__ val_w, const float* __restrict__ pos_w,
    unsigned* __restrict__ wsbits) {
  int t = blockIdx.x * blockDim.x + threadIdx.x;
  if (t >= TOT_DW) return;
  int r   = t >> 6;   // 0..131
  int w64 = t & 63;   // packed dword within row
  int h = w64 >> 5;   // which 1024-d half
  int w = w64 & 31;   // lane slot
  const float* row = (r < NLEV) ? (val_w + (size_t)r * DDIM)
                                : (pos_w + (size_t)(r - NLEV) * DDIM);
  unsigned m = 0u;
#pragma unroll
  for (int b = 0; b < 32; ++b) {
    float f = row[h * 1024 + b * 32 + w];
    m |= (f < 0.0f ? 1u : 0u) << b;   // sign bit of the bipolar entry
  }
  wsbits[t] = m;                       // layout: r*64 + w64
}

// ---------------------------------------------------------------------------
// Kernel 3: main encode. Each wave32 handles one (sample n, d-half h) pair:
//   lane c computes the level index, __shfl-broadcasts it, 32 XOR words
//   (val sign ^ pos sign), bit-sliced counters give per-d negative counts,
//   threshold >16 yields the sign, 32 coalesced b32 stores write the floats.
// Packed tables staged into LDS via the Tensor Data Mover (TENSORcnt path),
// with a redundant plain staging pass for descriptor-semantics safety.
// ---------------------------------------------------------------------------
__global__ __launch_bounds__(256) void hd_main_kernel(
    const float* __restrict__ feat, const float* __restrict__ mmx,
    const unsigned* __restrict__ wsbits, float* __restrict__ out) {
  __shared__ unsigned lds_bits[TOT_DW];

  // --- CDNA5 TDM: async 1-D tile copy global -> LDS (wave 0 issues it) ---
  if (threadIdx.x < 32) {
    unsigned long long ga = (unsigned long long)(uintptr_t)wsbits;
    unsigned ldsoff = (unsigned)(uintptr_t)(&lds_bits[0]);  // low 32b = LDS addr
    v4u g0;
    v8i g1 = {0, 0, 0, 0, 0, 0, 0, 0};
    v4i g2 = {0, 0, 0, 0};
    v4i g3 = {0, 0, 0, 0};
    v8i g4 = {0, 0, 0, 0, 0, 0, 0, 0};   // aux group (6-arg toolchain form)
    // Group0: count=1 | lds_addr | global_addr[56:0] | type=2 ("image")
    g0[0] = 1u;
    g0[1] = ldsoff;
    g0[2] = (unsigned)(ga & 0xFFFFFFFFu);
    g0[3] = (unsigned)((ga >> 32) & 0x01FFFFFFu) | (2u << 30);
    // Group1: wg_mask=0, data_size=2 (4B), tensor_dim0=tile_dim0=8448,
    //         tensor_dim1=tile_dim1=1, dim0_stride=8448
    g1[0] = (int)(2u << 16);                 // data_size = 4 bytes
    g1[1] = (int)((unsigned)TOT_DW << 16);   // tensor_dim0 bits[79:48] lo
    g1[2] = (int)(1u << 16);                 // tensor_dim1 = 1
    g1[3] = (int)((unsigned)TOT_DW << 16);   // tile_dim0 (16b)
    g1[4] = 1;                               // tile_dim1 = 1
    g1[5] = (int)TOT_DW;                     // tensor_dim0_stride lo
    __builtin_amdgcn_tensor_load_to_lds(g0, g1, g2, g3, g4, 0);
    __builtin_amdgcn_s_wait_tensorcnt(0);
  }
  // Redundant plain staging: guarantees LDS contents regardless of D# details
  for (int i = threadIdx.x; i < TOT_DW; i += 256) lds_bits[i] = wsbits[i];
  __syncthreads();

  const int lane = threadIdx.x & 31;
  const int wave = threadIdx.x >> 5;
  const int gwave = blockIdx.x * 8 + wave;
  const int nwaves = gridDim.x * 8;
  const unsigned* vb = lds_bits;            // 100 x 64 packed val rows
  const unsigned* pb = lds_bits + VAL_DW;   // 32 x 64 packed pos rows

  const float mn = mmx[lane];
  const float mx = mmx[32 + lane];
  const float denom = fmaxf(mx - mn, 1e-6f);

  for (int pair = gwave; pair < NSAMP * 2; pair += nwaves) {
    const int n = pair >> 1;
    const int h = pair & 1;

    // lane == channel: quantize exactly as the reference does:
    // f01 = (x - min) / denom  (IEEE div), idx = trunc(f01 * 99), clip
    float x = feat[n * NCH + lane];
    float f01 = (x - mn) / denom;
    int idxv = (int)(f01 * (float)(NLEV - 1));
    idxv = min(max(idxv, 0), NLEV - 1);

    const int off = h * 32 + lane;
    unsigned ones = 0, twos = 0, fours = 0, eights = 0, sixteens = 0, t32 = 0;
#pragma unroll
    for (int c = 0; c < 32; ++c) {
      int ic = __shfl(idxv, c);                         // broadcast idx[n,c]
      unsigned xw = vb[ic * DWPR + off] ^ pb[c * DWPR + off];
      // bit-sliced ripple add of one {0,1} plane into 6 counter planes
      unsigned cy = ones & xw;     ones ^= xw;
      unsigned cy2 = twos & cy;    twos ^= cy;
      unsigned cy3 = fours & cy2;  fours ^= cy2;
      unsigned cy4 = eights & cy3; eights ^= cy3;
      unsigned cy5 = sixteens & cy4; sixteens ^= cy4;
      t32 ^= cy5;
    }
    // neg_count > 16  <=>  hv < 0  (count==16 -> hv==0 -> +1, matches ref)
    unsigned gt16 = t32 | (sixteens & (eights | fours | twos | ones));

    // bit b of lane l  ->  d = h*1024 + b*32 + l : 32 fully coalesced stores
    unsigned* op = (unsigned*)(out + (size_t)n * DDIM + h * 1024 + lane);
#pragma unroll
    for (int b = 0; b < 32; ++b) {
      op[b * 32] = 0x3F800000u | (((gt16 >> b) & 1u) << 31);  // +/-1.0f
    }
  }
}

// ---------------------------------------------------------------------------
extern "C" void kernel_launch(void* const* d_in, const int* in_sizes, int n_in,
                              void* d_out, int out_size, void* d_ws, size_t ws_size,
                              hipStream_t stream) {
  (void)in_sizes; (void)n_in; (void)out_size; (void)ws_size;
  const float* feat  = (const float*)d_in[0];   // [4096, 32]
  const float* val_w = (const float*)d_in[1];   // [100, 2048]
  const float* pos_w = (const float*)d_in[2];   // [32, 2048]
  float* out = (float*)d_out;                   // [4096, 2048]

  float*    wsf    = (float*)d_ws;              // 64 floats: min/max
  unsigned* wsbits = (unsigned*)d_ws + 64;      // 8448 packed dwords

  hd_minmax_kernel<<<1, 1024, 0, stream>>>(feat, wsf);
  hd_pack_kernel<<<(TOT_DW + 255) / 256, 256, 0, stream>>>(val_w, pos_w, wsbits);
  hd_main_kernel<<<512, 256, 0, stream>>>(feat, wsf, wsbits, out);
}